// DisenEncoder_17978733101718
// MI455X (gfx1250) — compile-verified
//
#include <hip/hip_runtime.h>
#include <hip/hip_bf16.h>
#include <math.h>

typedef __attribute__((ext_vector_type(2))) float v2f;
typedef __attribute__((ext_vector_type(8))) float v8f;

#define IN_DIM 512
#define DOUT   128   // D = K * X_DIM = 2*64
#define NRM_EPS 1e-12f

// ---------------------------------------------------------------------------
// GEMM: h[N,128] = x[N,512] @ W[512,128] + b, using V_WMMA_F32_16X16X4_F32.
// Block = 256 threads = 8 waves. Block owns 16 rows; wave w owns col tile w.
// x tile (16x512 f32, contiguous in memory) staged into LDS with CDNA5
// async global->LDS copies (GLOBAL_LOAD_ASYNC_TO_LDS_B128, ASYNCcnt).
// ---------------------------------------------------------------------------
__global__ __launch_bounds__(256)
void gemm_bias_kernel(const float* __restrict__ x, const float* __restrict__ W,
                      const float* __restrict__ bias, float* __restrict__ h, int N)
{
    __shared__ float xs[16 * IN_DIM];   // 32 KB

    const int row0 = blockIdx.x * 16;
    const int tid  = threadIdx.x;

    // --- Async stage of the 16xIN_DIM x-tile into LDS (row-clamped tail) ---
    {
        // LDS byte address: generic shared address truncates to addr[31:0]
        unsigned xsb = (unsigned)(uintptr_t)&xs[0];
        #pragma unroll
        for (int i = 0; i < 8; ++i) {
            int idx = tid + i * 256;            // 2048 float4 total
            int r   = idx >> 7;                 // / (IN_DIM/4)
            int cc  = idx & 127;                // % (IN_DIM/4)
            int rmax = N - 1 - row0;
            int rr  = (r <= rmax) ? r : (rmax > 0 ? rmax : 0);
            const float4* g = ((const float4*)(x + (size_t)(row0 + rr) * IN_DIM)) + cc;
            unsigned l = xsb + (unsigned)idx * 16u;
            asm volatile("global_load_async_to_lds_b128 %0, %1, off"
                         :: "v"(l), "v"(g) : "memory");
        }
        // Drain this wave's outstanding async copies before the barrier.
        asm volatile("s_wait_asynccnt 0x0" ::: "memory");
    }
    __syncthreads();

    const int wave = tid >> 5;        // 0..7 -> column tile
    const int lane = tid & 31;
    const int mrow = lane & 15;       // A-matrix row / B,C,D column-in-tile
    const int half = lane >> 4;       // lane half selects K-pair / row offset
    const int col  = wave * 16 + mrow;
    const int koff = half * 2;

    v8f acc = {};
    for (int k0 = 0; k0 < IN_DIM; k0 += 4) {
        // A fragment (16x4): lane half 0 holds K = k0,k0+1; half 1 holds k0+2,k0+3
        float2 a2 = *(const float2*)&xs[mrow * IN_DIM + k0 + koff];
        v2f af; af.x = a2.x; af.y = a2.y;
        // B fragment (4x16), mirrored layout: rows k0+koff, k0+koff+1 at column `col`
        v2f bf;
        bf.x = W[(size_t)(k0 + koff)     * DOUT + col];
        bf.y = W[(size_t)(k0 + koff + 1) * DOUT + col];
        acc = __builtin_amdgcn_wmma_f32_16x16x4_f32(
            /*neg_a=*/false, af, /*neg_b=*/false, bf,
            /*c_mod=*/(short)0, acc, /*reuse_a=*/false, /*reuse_b=*/false);
    }

    const float bv = bias[col];
    if (row0 + 16 <= N) {
        // Full tile: unconditional stores (uniform branch, EXEC stays full).
        float* out = h + (size_t)(row0 + half * 8) * DOUT + col;
        #pragma unroll
        for (int j = 0; j < 8; ++j)
            out[(size_t)j * DOUT] = acc[j] + bv;
    } else {
        #pragma unroll
        for (int j = 0; j < 8; ++j) {
            int grow = row0 + j + half * 8;
            if (grow < N) h[(size_t)grow * DOUT + col] = acc[j] + bv;
        }
    }
}

// ---------------------------------------------------------------------------
// Per-channel L2 normalize: one wave per node, one float4 per lane.
// Lanes 0..15 = channel 0 (cols 0..63), lanes 16..31 = channel 1.
// ---------------------------------------------------------------------------
__global__ __launch_bounds__(256)
void normalize_kernel(float* __restrict__ c, int N)
{
    int wid  = (int)((blockIdx.x * (unsigned)blockDim.x + threadIdx.x) >> 5);
    int lane = threadIdx.x & 31;
    if (wid >= N) return;

    float4* c4 = (float4*)c;
    size_t idx = (size_t)wid * 32 + lane;
    float4 v = c4[idx];
    float ss = v.x * v.x + v.y * v.y + v.z * v.z + v.w * v.w;
    // reduce over the 16 lanes of this channel half
    ss += __shfl_xor(ss, 1, 32);
    ss += __shfl_xor(ss, 2, 32);
    ss += __shfl_xor(ss, 4, 32);
    ss += __shfl_xor(ss, 8, 32);
    float scale = 1.0f / fmaxf(sqrtf(ss), NRM_EPS);
    v.x *= scale; v.y *= scale; v.z *= scale; v.w *= scale;
    c4[idx] = v;
}

// ---------------------------------------------------------------------------
// One routing iteration: per edge e: z = h[src], ct = cprev[trg],
// p = softmax_k(<z_k, ct_k>), cnext[trg] += p_k * z_k  (atomic, L2 resident).
// One wave per edge; lane j owns float4 j of the 128-float row.
// ---------------------------------------------------------------------------
__global__ __launch_bounds__(256)
void route_kernel(const float* __restrict__ h, const float* __restrict__ cprev,
                  float* __restrict__ cnext, const long long* __restrict__ st,
                  int M)
{
    long long wid = ((long long)blockIdx.x * blockDim.x + threadIdx.x) >> 5;
    int lane = threadIdx.x & 31;
    if (wid >= M) return;

    int src = (int)st[wid];
    int trg = (int)st[(size_t)M + wid];

    const float4* h4 = (const float4*)h;
    const float4* c4 = (const float4*)cprev;
    float4 z = h4[(size_t)src * 32 + lane];
    float4 t = c4[(size_t)trg * 32 + lane];

    float d = z.x * t.x + z.y * t.y + z.z * t.z + z.w * t.w;
    d += __shfl_xor(d, 1, 32);
    d += __shfl_xor(d, 2, 32);
    d += __shfl_xor(d, 4, 32);
    d += __shfl_xor(d, 8, 32);                 // channel dot in each half
    float other = __shfl_xor(d, 16, 32);       // the other channel's dot
    float mx = fmaxf(d, other);
    float e0 = __expf(d - mx), e1 = __expf(other - mx);
    float p = e0 / (e0 + e1);                  // softmax weight of MY channel

    float* dst = cnext + (size_t)trg * DOUT + lane * 4;
    atomicAdd(dst + 0, p * z.x);
    atomicAdd(dst + 1, p * z.y);
    atomicAdd(dst + 2, p * z.z);
    atomicAdd(dst + 3, p * z.w);
}

// ---------------------------------------------------------------------------
extern "C" void kernel_launch(void* const* d_in, const int* in_sizes, int n_in,
                              void* d_out, int out_size, void* d_ws, size_t ws_size,
                              hipStream_t stream) {
    const float*     x    = (const float*)d_in[0];
    const long long* st   = (const long long*)d_in[1];  // int64 [2, M]
    const float*     W    = (const float*)d_in[2];
    const float*     bias = (const float*)d_in[3];

    const int N = in_sizes[0] / IN_DIM;
    const int M = in_sizes[1] / 2;

    float* h    = (float*)d_ws;                       // [N,128] normalized features (z source)
    float* buf1 = h + (size_t)N * DOUT;               // scratch c
    float* cout = (float*)d_out;                      // final c

    const size_t cbytes = (size_t)N * DOUT * sizeof(float);

    dim3 blk(256);
    dim3 gemm_grid((N + 15) / 16);
    dim3 node_grid((unsigned)(((long long)N * 32 + 255) / 256));
    dim3 edge_grid((unsigned)(((long long)M * 32 + 255) / 256));

    // h = normalize(x @ W + b)
    gemm_bias_kernel<<<gemm_grid, blk, 0, stream>>>(x, W, bias, h, N);
    normalize_kernel<<<node_grid, blk, 0, stream>>>(h, N);

    // iter 1: prev = h, next = d_out
    hipMemcpyAsync(cout, h, cbytes, hipMemcpyDeviceToDevice, stream);
    route_kernel<<<edge_grid, blk, 0, stream>>>(h, h, cout, st, M);
    normalize_kernel<<<node_grid, blk, 0, stream>>>(cout, N);

    // iter 2: prev = d_out, next = buf1
    hipMemcpyAsync(buf1, cout, cbytes, hipMemcpyDeviceToDevice, stream);
    route_kernel<<<edge_grid, blk, 0, stream>>>(h, cout, buf1, st, M);
    normalize_kernel<<<node_grid, blk, 0, stream>>>(buf1, N);

    // iter 3: prev = buf1, next = d_out (final result lands in d_out)
    hipMemcpyAsync(cout, buf1, cbytes, hipMemcpyDeviceToDevice, stream);
    route_kernel<<<edge_grid, blk, 0, stream>>>(h, buf1, cout, st, M);
    normalize_kernel<<<node_grid, blk, 0, stream>>>(cout, N);
}